// Hy_LTra_43155831390749
// MI455X (gfx1250) — compile-verified
//
#include <hip/hip_runtime.h>
#include <math.h>

// ---------------- model constants ----------------
#define B_    4
#define T_    128
#define M_    120
#define D_    128
#define NH_   8
#define DH_   16
#define FF_   2048
#define NL_   2
#define HID_  100
#define CD_   2
#define EPS_  1e-5f
#define LDW   132     // LDS row stride (floats): %64 = 4 -> conflict-free, 8B aligned
#define CROWS 130     // rows for conv-staged buffers (+1 shift, +1 slack)

typedef float v2f __attribute__((ext_vector_type(2)));
typedef float v8f __attribute__((ext_vector_type(8)));

// ---------------- fp32 WMMA 16x16x4 ----------------
__device__ __forceinline__ v8f wmma4(v2f a, v2f b, v8f c) {
  // D = A(16x4) * B(4x16) + C(16x16), fp32
  return __builtin_amdgcn_wmma_f32_16x16x4_f32(false, a, false, b, (short)0, c, false, false);
}

// 16x16 output tile: A (row-major, lda) rows r0..r0+15,
// B[k][n] = Bm[(n0+n)*ldb + k] (i.e. y = A @ Bm^T), K multiple of 4.
__device__ __forceinline__ v8f gemm16(const float* A, int r0, int lda,
                                      const float* Bm, int n0, int ldb,
                                      int K, v8f acc, int lm, int hf) {
  const float* arow = A + (size_t)(r0 + lm) * lda + 2 * hf;
  const float* brow = Bm + (size_t)(n0 + lm) * ldb + 2 * hf;
  for (int k = 0; k < K; k += 4) {
    v2f a = *(const v2f*)(arow + k);
    v2f b = *(const v2f*)(brow + k);
    acc = wmma4(a, b, acc);
  }
  return acc;
}

// O = S @ V : B[k][n] = V[k*ldv + n0+n] (V row-major, not transposed)
__device__ __forceinline__ v8f gemm16_sv(const float* S, int r0, int lda,
                                         const float* V, int n0, int ldv,
                                         int K, v8f acc, int lm, int hf) {
  const float* arow = S + (size_t)(r0 + lm) * lda + 2 * hf;
  for (int k = 0; k < K; k += 4) {
    v2f a = *(const v2f*)(arow + k);
    v2f b;
    b.x = V[(size_t)(k + 2 * hf) * ldv + n0 + lm];
    b.y = V[(size_t)(k + 2 * hf + 1) * ldv + n0 + lm];
    acc = wmma4(a, b, acc);
  }
  return acc;
}

__device__ __forceinline__ void store_tile(float* Out, int r0, int c0, int ld,
                                           v8f acc, int lm, int hf) {
#pragma unroll
  for (int r = 0; r < 8; ++r)
    Out[(size_t)(r0 + r + 8 * hf) * ld + c0 + lm] = acc[r];
}

__device__ __forceinline__ v8f load_tile(const float* In, int r0, int c0, int ld,
                                         int lm, int hf) {
  v8f c;
#pragma unroll
  for (int r = 0; r < 8; ++r)
    c[r] = In[(size_t)(r0 + r + 8 * hf) * ld + c0 + lm];
  return c;
}

// LayerNorm over 128 cols of (H + Acc), result into H. One thread per row.
__device__ __forceinline__ void ln_rows(float* H, const float* Acc,
                                        const float* g, const float* be,
                                        int N, int tid) {
  for (int p = tid; p < N; p += 256) {
    float mu = 0.f;
    for (int d = 0; d < 128; ++d) mu += H[p * LDW + d] + Acc[p * LDW + d];
    mu *= (1.f / 128.f);
    float var = 0.f;
    for (int d = 0; d < 128; ++d) {
      float t = H[p * LDW + d] + Acc[p * LDW + d] - mu;
      var += t * t;
    }
    var *= (1.f / 128.f);
    float inv = rsqrtf(var + EPS_);
    for (int d = 0; d < 128; ++d) {
      float t = H[p * LDW + d] + Acc[p * LDW + d];
      H[p * LDW + d] = (t - mu) * inv * g[d] + be[d];
    }
  }
}

struct EncP {
  const float *x;
  const float *w1t, *c1b, *bn1g, *bn1b, *bn1m, *bn1v;   // w1t: (3,128,120)
  const float *w2t, *c2b, *bn2g, *bn2b, *bn2m, *bn2v;   // w2t: (3,128,128)
  const float *emw, *emb;
  const float *aiw, *aib, *aow, *aob, *l1g, *l1b;
  const float *f1w, *f1b, *f2w, *f2b, *l2g, *l2b;
  const float *ow, *ob;
  float *Aout; // (B,T) anomaly scores
};

// transpose conv weights: (Cout,Cin,3) -> (3,Cout,Cin) so WMMA B-operand
// loads are consecutive in Cin.
__global__ void tw_kernel(const float* c1w, const float* c2w,
                          float* W1t, float* W2t) {
  int idx = blockIdx.x * blockDim.x + threadIdx.x;
  if (idx < 3 * 128 * M_) {
    int k = idx / (128 * M_);
    int rem = idx - k * (128 * M_);
    int co = rem / M_, ci = rem - co * M_;
    W1t[idx] = c1w[(size_t)(co * M_ + ci) * 3 + k];
  }
  int j = idx - 3 * 128 * M_;
  if (j >= 0 && j < 3 * 128 * 128) {
    int k = j / (128 * 128);
    int rem = j - k * (128 * 128);
    int co = rem >> 7, ci = rem & 127;
    W2t[j] = c2w[(size_t)((co << 7) + ci) * 3 + k];
  }
}

// One block per (L, b): full prefix encoder (positions 0..L-1) in LDS,
// writes Aout[b*T + L] = mean((h[L-1]@ow^T + ob - x[b,L])^2).
__launch_bounds__(256)
__global__ void enc_kernel(EncP P) {
  __shared__ float Hs[128 * LDW];    // h
  __shared__ float As[CROWS * LDW];  // x-stage(+1) / conv2 / scores / FFN acc
  __shared__ float Bs[CROWS * LDW];  // conv1(+1) / attn-out / FFN tmp
  __shared__ float Qh[128 * 20];
  __shared__ float Kh[128 * 20];
  __shared__ float Vh[128 * 20];
  __shared__ float red[256];

  const int tid = threadIdx.x;
  const int lane = tid & 31, wv = tid >> 5;
  const int lm = lane & 15, hf = lane >> 4;
  const int b = blockIdx.x & 3;
  const int L = (blockIdx.x >> 2) + 1;
  const int N = L;
  const int NT = (N + 15) >> 4;
  const int NP = NT * 16;

  // zero LDS so pad rows/cols are well-defined everywhere
  for (int i = tid; i < CROWS * LDW; i += 256) {
    As[i] = 0.f; Bs[i] = 0.f;
    if (i < 128 * LDW) Hs[i] = 0.f;
  }
  for (int i = tid; i < 128 * 20; i += 256) { Qh[i] = 0.f; Kh[i] = 0.f; Vh[i] = 0.f; }
  __syncthreads();

  // stage x prefix at +1 row offset (row 0 = left pad, rows >= N+1 stay zero)
  const float* xb = P.x + (size_t)b * T_ * M_;
  for (int i = tid; i < N * M_; i += 256) {
    int p = i / M_, c = i - p * M_;
    As[(p + 1) * LDW + c] = xb[p * M_ + c];
  }
  __syncthreads();

  // conv1 + bn1 + relu -> Bs (stored at +1 row offset), via 3 shifted GEMMs
  for (int job = wv; job < NT * 8; job += 8) {
    int rt = job >> 3, ct = job & 7;
    v8f acc = {0.f, 0.f, 0.f, 0.f, 0.f, 0.f, 0.f, 0.f};
    for (int k = 0; k < 3; ++k)
      acc = gemm16(As, rt * 16 + k, LDW, P.w1t + (size_t)k * 128 * M_,
                   ct * 16, M_, M_, acc, lm, hf);
#pragma unroll
    for (int r = 0; r < 8; ++r) {
      int p = rt * 16 + r + 8 * hf;
      int co = ct * 16 + lm;
      float s = P.bn1g[co] * rsqrtf(P.bn1v[co] + EPS_);
      float v = (acc[r] + P.c1b[co] - P.bn1m[co]) * s + P.bn1b[co];
      v = v > 0.f ? v : 0.f;
      Bs[(p + 1) * LDW + co] = (p < N) ? v : 0.f;  // keep mask invariant
    }
  }
  __syncthreads();

  // conv2 + bn2 + relu -> As (0-based rows)
  for (int job = wv; job < NT * 8; job += 8) {
    int rt = job >> 3, ct = job & 7;
    v8f acc = {0.f, 0.f, 0.f, 0.f, 0.f, 0.f, 0.f, 0.f};
    for (int k = 0; k < 3; ++k)
      acc = gemm16(Bs, rt * 16 + k, LDW, P.w2t + (size_t)k * 128 * 128,
                   ct * 16, 128, 128, acc, lm, hf);
#pragma unroll
    for (int r = 0; r < 8; ++r) {
      int p = rt * 16 + r + 8 * hf;
      int co = ct * 16 + lm;
      float s = P.bn2g[co] * rsqrtf(P.bn2v[co] + EPS_);
      float v = (acc[r] + P.c2b[co] - P.bn2m[co]) * s + P.bn2b[co];
      v = v > 0.f ? v : 0.f;
      As[p * LDW + co] = (p < N) ? v : 0.f;
    }
  }
  __syncthreads();

  // embed: Hs = As @ emw^T + emb + PE
  for (int job = wv; job < NT * 8; job += 8) {
    int rt = job >> 3, ct = job & 7;
    v8f acc = {0.f, 0.f, 0.f, 0.f, 0.f, 0.f, 0.f, 0.f};
    acc = gemm16(As, rt * 16, LDW, P.emw, ct * 16, 128, 128, acc, lm, hf);
#pragma unroll
    for (int r = 0; r < 8; ++r) {
      int p = rt * 16 + r + 8 * hf;
      int d = ct * 16 + lm;
      int i2 = d >> 1;
      float div = expf((float)(2 * i2) * (-logf(10000.f) / 128.f));
      float ang = (float)p * div;
      float pe = (d & 1) ? cosf(ang) : sinf(ang);
      Hs[p * LDW + d] = acc[r] + P.emb[d] + pe;
    }
  }
  __syncthreads();

  // -------- transformer layers --------
  for (int l = 0; l < NL_; ++l) {
    const float* Wi = P.aiw + (size_t)l * 384 * 128;
    const float* bi = P.aib + l * 384;
    const float* Wo = P.aow + (size_t)l * 128 * 128;
    const float* bo = P.aob + l * 128;

    for (int hd = 0; hd < NH_; ++hd) {
      // per-head Q,K,V  (q pre-scaled by 1/sqrt(dh)=0.25)
      for (int job = wv; job < 3 * NT; job += 8) {
        int which = job / NT, rt = job - which * NT;
        const float* W = Wi + (size_t)(which * 128 + hd * 16) * 128;
        const float* bb = bi + which * 128 + hd * 16;
        float* Dst = (which == 0) ? Qh : ((which == 1) ? Kh : Vh);
        v8f acc = {0.f, 0.f, 0.f, 0.f, 0.f, 0.f, 0.f, 0.f};
        acc = gemm16(Hs, rt * 16, LDW, W, 0, 128, 128, acc, lm, hf);
        float scal = (which == 0) ? 0.25f : 1.f;
#pragma unroll
        for (int r = 0; r < 8; ++r) {
          int p = rt * 16 + r + 8 * hf;
          Dst[p * 20 + lm] = (acc[r] + bb[lm]) * scal;
        }
      }
      __syncthreads();

      // scores S = Q @ K^T, causal mask -> As
      for (int job = wv; job < NT * NT; job += 8) {
        int rt = job / NT, ct = job - rt * NT;
        v8f acc = {0.f, 0.f, 0.f, 0.f, 0.f, 0.f, 0.f, 0.f};
        acc = gemm16(Qh, rt * 16, 20, Kh, ct * 16, 20, 16, acc, lm, hf);
#pragma unroll
        for (int r = 0; r < 8; ++r) {
          int q = rt * 16 + r + 8 * hf;
          int k = ct * 16 + lm;
          As[q * LDW + k] = (k > q || k >= N) ? -1e30f : acc[r];
        }
      }
      __syncthreads();

      // softmax rows (valid rows only; masked cols -> 0)
      for (int p = tid; p < N; p += 256) {
        float mx = -1e30f;
        for (int j = 0; j < NP; ++j) mx = fmaxf(mx, As[p * LDW + j]);
        float sm = 0.f;
        for (int j = 0; j < NP; ++j) {
          float e = __expf(As[p * LDW + j] - mx);
          As[p * LDW + j] = e;
          sm += e;
        }
        float inv = 1.f / sm;
        for (int j = 0; j < NP; ++j) As[p * LDW + j] *= inv;
      }
      __syncthreads();

      // O_head = S @ V -> Bs[:, hd*16 .. hd*16+15]
      for (int job = wv; job < NT; job += 8) {
        v8f acc = {0.f, 0.f, 0.f, 0.f, 0.f, 0.f, 0.f, 0.f};
        acc = gemm16_sv(As, job * 16, LDW, Vh, 0, 20, NP, acc, lm, hf);
#pragma unroll
        for (int r = 0; r < 8; ++r) {
          int p = job * 16 + r + 8 * hf;
          Bs[p * LDW + hd * 16 + lm] = acc[r];
        }
      }
      __syncthreads();
    }

    // attn out-proj: As = Bs @ Wo^T + bo
    for (int job = wv; job < NT * 8; job += 8) {
      int rt = job >> 3, ct = job & 7;
      v8f acc = {0.f, 0.f, 0.f, 0.f, 0.f, 0.f, 0.f, 0.f};
      acc = gemm16(Bs, rt * 16, LDW, Wo, ct * 16, 128, 128, acc, lm, hf);
#pragma unroll
      for (int r = 0; r < 8; ++r) {
        int p = rt * 16 + r + 8 * hf;
        int d = ct * 16 + lm;
        As[p * LDW + d] = acc[r] + bo[d];
      }
    }
    __syncthreads();

    ln_rows(Hs, As, P.l1g + l * 128, P.l1b + l * 128, N, tid);
    __syncthreads();

    // FFN: As(acc) = relu(Hs@W1^T+b1)@W2^T, chunked over FF
    const float* W1 = P.f1w + (size_t)l * FF_ * 128;
    const float* b1 = P.f1b + l * FF_;
    const float* W2 = P.f2w + (size_t)l * 128 * FF_;
    for (int i = tid; i < NP * 128; i += 256) As[(i >> 7) * LDW + (i & 127)] = 0.f;
    __syncthreads();
    for (int c = 0; c < FF_ / 128; ++c) {
      for (int job = wv; job < NT * 8; job += 8) {
        int rt = job >> 3, ct = job & 7;
        v8f acc = {0.f, 0.f, 0.f, 0.f, 0.f, 0.f, 0.f, 0.f};
        acc = gemm16(Hs, rt * 16, LDW, W1 + (size_t)(c * 128) * 128, ct * 16, 128,
                     128, acc, lm, hf);
#pragma unroll
        for (int r = 0; r < 8; ++r) {
          int p = rt * 16 + r + 8 * hf;
          int j = ct * 16 + lm;
          float t = acc[r] + b1[c * 128 + j];
          Bs[p * LDW + j] = t > 0.f ? t : 0.f;
        }
      }
      __syncthreads();
      for (int job = wv; job < NT * 8; job += 8) {
        int rt = job >> 3, ct = job & 7;
        v8f acc = load_tile(As, rt * 16, ct * 16, LDW, lm, hf);
        acc = gemm16(Bs, rt * 16, LDW, W2 + c * 128, ct * 16, FF_, 128, acc, lm, hf);
        store_tile(As, rt * 16, ct * 16, LDW, acc, lm, hf);
      }
      __syncthreads();
    }
    // add ff2 bias then LN2
    {
      const float* b2 = P.f2b + l * 128;
      for (int i = tid; i < N * 128; i += 256)
        As[(i >> 7) * LDW + (i & 127)] += b2[i & 127];
    }
    __syncthreads();
    ln_rows(Hs, As, P.l2g + l * 128, P.l2b + l * 128, N, tid);
    __syncthreads();
  }

  // pred = Hs[L-1] @ ow^T + ob ; anomaly = mean((pred - x[b,L])^2)
  {
    const int p = N - 1;
    float part = 0.f;
    for (int m = tid; m < M_; m += 256) {
      float acc = P.ob[m];
      const float* w = P.ow + (size_t)m * 128;
      for (int d = 0; d < 128; ++d) acc += Hs[p * LDW + d] * w[d];
      float e = acc - xb[(size_t)L * M_ + m];
      part += e * e;
    }
    red[tid] = part;
    __syncthreads();
    for (int s = 128; s > 0; s >>= 1) {
      if (tid < s) red[tid] += red[tid + s];
      __syncthreads();
    }
    if (tid == 0) P.Aout[b * T_ + L] = red[0] * (1.f / M_);
  }
}

// gating + build LSTM input X = [rssi*g, prev] (B,T,122); also emit gating output
__global__ void gate_kernel(const float* rssi, const float* prev, const float* A,
                            float* X, float* gout) {
  int idx = blockIdx.x * blockDim.x + threadIdx.x;
  if (idx >= B_ * T_) return;
  int bb = idx / T_, t = idx - bb * T_;
  float a = (t == 0) ? 0.f : A[bb * T_ + t];
  float g = expf(-0.5f * a);
  gout[bb * T_ + t] = g;
  float* xr = X + (size_t)idx * (M_ + CD_);
  const float* rr = rssi + (size_t)idx * M_;
  for (int c = 0; c < M_; ++c) xr[c] = rr[c] * g;
  xr[M_ + 0] = prev[(size_t)idx * CD_ + 0];
  xr[M_ + 1] = prev[(size_t)idx * CD_ + 1];
}

// XG = Xin @ Wih^T + bih + bhh   (rows = B*T, cols = 400)
__global__ void xg_kernel(const float* Xin, int ldx, const float* Wih,
                          const float* bih, const float* bhh, float* XG) {
  int idx = blockIdx.x * blockDim.x + threadIdx.x;
  if (idx >= B_ * T_ * 4 * HID_) return;
  int row = idx / (4 * HID_), j = idx - row * (4 * HID_);
  const float* x = Xin + (size_t)row * ldx;
  const float* w = Wih + (size_t)j * ldx;
  float acc = bih[j] + bhh[j];
  for (int k = 0; k < ldx; ++k) acc += x[k] * w[k];
  XG[(size_t)row * (4 * HID_) + j] = acc;
}

// one block per batch: sequential LSTM; optionally fused FC head
__launch_bounds__(256)
__global__ void lstm_kernel(const float* XG, const float* Whh, float* Hseq,
                            const float* fcw, const float* fcb, float* traj) {
  __shared__ float h[HID_], c[HID_], g[4 * HID_];
  int b = blockIdx.x, tid = threadIdx.x;
  for (int i = tid; i < HID_; i += 256) { h[i] = 0.f; c[i] = 0.f; }
  __syncthreads();
  for (int t = 0; t < T_; ++t) {
    const float* xg = XG + ((size_t)b * T_ + t) * (4 * HID_);
    for (int j = tid; j < 4 * HID_; j += 256) {
      const float* w = Whh + (size_t)j * HID_;
      float acc = xg[j];
      for (int k = 0; k < HID_; ++k) acc += h[k] * w[k];
      g[j] = acc;
    }
    __syncthreads();
    for (int k = tid; k < HID_; k += 256) {
      float i_ = 1.f / (1.f + expf(-g[k]));
      float f_ = 1.f / (1.f + expf(-g[HID_ + k]));
      float gg = tanhf(g[2 * HID_ + k]);
      float o_ = 1.f / (1.f + expf(-g[3 * HID_ + k]));
      float cn = f_ * c[k] + i_ * gg;
      c[k] = cn;
      h[k] = o_ * tanhf(cn);
    }
    __syncthreads();
    for (int k = tid; k < HID_; k += 256) Hseq[((size_t)b * T_ + t) * HID_ + k] = h[k];
    if (fcw != nullptr && tid < CD_) {
      float acc = fcb[tid];
      for (int k = 0; k < HID_; ++k) acc += h[k] * fcw[tid * HID_ + k];
      traj[((size_t)b * T_ + t) * CD_ + tid] = acc;
    }
    __syncthreads();
  }
}

enum {
  IN_RSSI = 0, IN_PREV, IN_C1W, IN_C1B, IN_BN1G, IN_BN1B, IN_BN1M, IN_BN1V,
  IN_C2W, IN_C2B, IN_BN2G, IN_BN2B, IN_BN2M, IN_BN2V,
  IN_EMW, IN_EMB, IN_AIW, IN_AIB, IN_AOW, IN_AOB, IN_LN1G, IN_LN1B,
  IN_FF1W, IN_FF1B, IN_FF2W, IN_FF2B, IN_LN2G, IN_LN2B, IN_OW, IN_OB,
  IN_L0WIH, IN_L0WHH, IN_L0BIH, IN_L0BHH, IN_L1WIH, IN_L1WHH, IN_L1BIH, IN_L1BHH,
  IN_FCW, IN_FCB
};

extern "C" void kernel_launch(void* const* d_in, const int* in_sizes, int n_in,
                              void* d_out, int out_size, void* d_ws, size_t ws_size,
                              hipStream_t stream) {
  (void)in_sizes; (void)n_in; (void)out_size; (void)ws_size;
  #define IN(i) ((const float*)d_in[(i)])

  float* ws = (float*)d_ws;
  float* Aanom = ws;                        // 512
  float* X     = Aanom + B_ * T_;           // 4*128*122 = 62464
  float* XG    = X + B_ * T_ * (M_ + CD_);  // 4*128*400 = 204800
  float* H0    = XG + B_ * T_ * 4 * HID_;   // 51200
  float* H1    = H0 + B_ * T_ * HID_;       // 51200
  float* W1t   = H1 + B_ * T_ * HID_;       // 3*128*120 = 46080
  float* W2t   = W1t + 3 * 128 * M_;        // 3*128*128 = 49152

  float* traj = (float*)d_out;                  // (B,T,2) = 1024 floats
  float* gout = (float*)d_out + B_ * T_ * CD_;  // (B,T,1) = 512 floats

  EncP P;
  P.x = IN(IN_RSSI);
  P.w1t = W1t;         P.c1b = IN(IN_C1B);
  P.bn1g = IN(IN_BN1G); P.bn1b = IN(IN_BN1B); P.bn1m = IN(IN_BN1M); P.bn1v = IN(IN_BN1V);
  P.w2t = W2t;         P.c2b = IN(IN_C2B);
  P.bn2g = IN(IN_BN2G); P.bn2b = IN(IN_BN2B); P.bn2m = IN(IN_BN2M); P.bn2v = IN(IN_BN2V);
  P.emw = IN(IN_EMW);  P.emb = IN(IN_EMB);
  P.aiw = IN(IN_AIW);  P.aib = IN(IN_AIB);
  P.aow = IN(IN_AOW);  P.aob = IN(IN_AOB);
  P.l1g = IN(IN_LN1G); P.l1b = IN(IN_LN1B);
  P.f1w = IN(IN_FF1W); P.f1b = IN(IN_FF1B);
  P.f2w = IN(IN_FF2W); P.f2b = IN(IN_FF2B);
  P.l2g = IN(IN_LN2G); P.l2b = IN(IN_LN2B);
  P.ow = IN(IN_OW);    P.ob = IN(IN_OB);
  P.Aout = Aanom;

  tw_kernel<<<dim3((3 * 128 * M_ + 3 * 128 * 128 + 255) / 256), dim3(256), 0, stream>>>(
      IN(IN_C1W), IN(IN_C2W), W1t, W2t);
  enc_kernel<<<dim3((T_ - 1) * B_), dim3(256), 0, stream>>>(P);
  gate_kernel<<<dim3(2), dim3(256), 0, stream>>>(IN(IN_RSSI), IN(IN_PREV), Aanom, X, gout);
  xg_kernel<<<dim3(800), dim3(256), 0, stream>>>(X, M_ + CD_, IN(IN_L0WIH),
                                                 IN(IN_L0BIH), IN(IN_L0BHH), XG);
  lstm_kernel<<<dim3(B_), dim3(256), 0, stream>>>(XG, IN(IN_L0WHH), H0,
                                                  nullptr, nullptr, nullptr);
  xg_kernel<<<dim3(800), dim3(256), 0, stream>>>(H0, HID_, IN(IN_L1WIH),
                                                 IN(IN_L1BIH), IN(IN_L1BHH), XG);
  lstm_kernel<<<dim3(B_), dim3(256), 0, stream>>>(XG, IN(IN_L1WHH), H1,
                                                  IN(IN_FCW), IN(IN_FCB), traj);
  #undef IN
}